// DatasetGraphGCN_20684562498266
// MI455X (gfx1250) — compile-verified
//
#include <hip/hip_runtime.h>
#include <hip/hip_bf16.h>

typedef __attribute__((ext_vector_type(2))) float v2f;
typedef __attribute__((ext_vector_type(8))) float v8f;

#define IN_F  128
#define HID   1024
#define OUT_F 64
#define MTILE 32          // node rows per block (2 x 16-row WMMA tiles)

// ---------------- degree / norm kernels ----------------

__global__ void k_deg_init(float* deg, int n) {
    int i = blockIdx.x * blockDim.x + threadIdx.x;
    if (i < n) deg[i] = 1.0f;   // self-loop weight
}

__global__ void k_deg_acc(const int* __restrict__ dst, const float* __restrict__ ew,
                          float* deg, int e) {
    int i = blockIdx.x * blockDim.x + threadIdx.x;
    if (i < e) atomicAdd(&deg[dst[i]], ew[i]);
}

__global__ void k_dinv(float* deg, int n) {
    int i = blockIdx.x * blockDim.x + threadIdx.x;
    if (i < n) {
        float d = deg[i];
        deg[i] = d > 0.0f ? rsqrtf(d) : 0.0f;   // deg buffer becomes dinv
    }
}

// agg1[i,:] = dinv[i]^2 * x[i,:]   (self-loop contribution)
__global__ void k_agg_init(const float* __restrict__ x, const float* __restrict__ dinv,
                           float* __restrict__ agg, int n) {
    long long i = (long long)blockIdx.x * blockDim.x + threadIdx.x;
    long long total = (long long)n * IN_F;
    if (i < total) {
        int node = (int)(i >> 7);   // /128
        float dv = dinv[node];
        agg[i] = dv * dv * x[i];
    }
}

// agg1[d,:] += dinv[s]*w*dinv[d] * x[s,:]   (two edges per 256-thread block)
__global__ void k_edge_agg1(const int* __restrict__ src, const int* __restrict__ dst,
                            const float* __restrict__ ew, const float* __restrict__ dinv,
                            const float* __restrict__ x, float* __restrict__ agg, int e) {
    int eid = blockIdx.x * 2 + (threadIdx.x >> 7);
    int f = threadIdx.x & 127;
    if (eid < e) {
        int s = src[eid], d = dst[eid];
        float norm = dinv[s] * ew[eid] * dinv[d];
        atomicAdd(&agg[(long long)d * IN_F + f], norm * x[(long long)s * IN_F + f]);
    }
}

// out[i,:] = dinv[i]^2 * t[i,:] + b2    (self-loop + bias)
__global__ void k_out_init(const float* __restrict__ t, const float* __restrict__ dinv,
                           const float* __restrict__ b2, float* __restrict__ out, int n) {
    long long i = (long long)blockIdx.x * blockDim.x + threadIdx.x;
    long long total = (long long)n * OUT_F;
    if (i < total) {
        int node = (int)(i >> 6);   // /64
        int f = (int)(i & 63);
        float dv = dinv[node];
        out[i] = dv * dv * t[i] + b2[f];
    }
}

// out[d,:] += dinv[s]*w*dinv[d] * t[s,:]   (four edges per 256-thread block)
__global__ void k_edge_agg2(const int* __restrict__ src, const int* __restrict__ dst,
                            const float* __restrict__ ew, const float* __restrict__ dinv,
                            const float* __restrict__ t, float* __restrict__ out, int e) {
    int eid = blockIdx.x * 4 + (threadIdx.x >> 6);
    int f = threadIdx.x & 63;
    if (eid < e) {
        int s = src[eid], d = dst[eid];
        float norm = dinv[s] * ew[eid] * dinv[d];
        atomicAdd(&out[(long long)d * OUT_F + f], norm * t[(long long)s * OUT_F + f]);
    }
}

// ---------------- fused WMMA GEMM1 -> relu -> GEMM2 ----------------
// Per block: 32 node rows (two 16-row WMMA M-tiles; B operands reused 2x).
//   L1: [32,128] @ W1[128,1024] (+b1, relu) -> LDS [32,1024]
//   L2: LDS [32,1024] @ W2[1024,64] -> t [32,64]
// f32 WMMA 16x16x4; A/B layouts per CDNA5 ISA 7.12.2.
__global__ __launch_bounds__(256) void k_fused_gemm(
    const float* __restrict__ agg1,   // [N,128]
    const float* __restrict__ W1,     // [128,1024] row-major
    const float* __restrict__ b1,     // [1024]
    const float* __restrict__ W2,     // [1024,64] row-major
    float* __restrict__ t,            // [N,64]
    int n_nodes)
{
    __shared__ float sA[MTILE][IN_F + 1];   //  16.5 KB (padded vs bank conflicts)
    __shared__ float sH[MTILE][HID + 1];    // 131.2 KB

    const int row0 = blockIdx.x * MTILE;
    const int tid  = threadIdx.x;
    const int wave = tid >> 5;              // 0..7 (wave32)
    const int lane = tid & 31;
    const int m    = lane & 15;             // matrix row (A) / column (B,C)
    const int hi   = lane >> 4;             // K-half select for A/B, M-half for C

    // cooperative, coalesced load of the 32x128 A tile
    for (int i = tid; i < MTILE * IN_F; i += 256) {
        int r = i >> 7;
        int c = i & 127;
        int rr = row0 + r;
        sA[r][c] = (rr < n_nodes) ? agg1[(long long)rr * IN_F + c] : 0.0f;
    }
    __syncthreads();

    // ---- Layer 1: each wave covers 8 n-tiles; each B load feeds 2 M-tiles ----
    for (int j = 0; j < 8; ++j) {
        int ncol = (wave * 8 + j) * 16 + m;      // output column (0..1023)
        v8f acc0 = {};                           // rows 0..15
        v8f acc1 = {};                           // rows 16..31
        #pragma unroll 4
        for (int kk = 0; kk < IN_F / 4; ++kk) {
            int k0 = kk * 4 + 2 * hi;
            v2f a0, a1, b;
            a0.x = sA[m][k0];       a0.y = sA[m][k0 + 1];
            a1.x = sA[16 + m][k0];  a1.y = sA[16 + m][k0 + 1];
            b.x  = W1[(long long)k0 * HID + ncol];
            b.y  = W1[(long long)(k0 + 1) * HID + ncol];
            acc0 = __builtin_amdgcn_wmma_f32_16x16x4_f32(
                false, a0, false, b, (short)0, acc0, false, false);
            acc1 = __builtin_amdgcn_wmma_f32_16x16x4_f32(
                false, a1, false, b, (short)0, acc1, false, false);
        }
        float bias = b1[ncol];
        #pragma unroll
        for (int r = 0; r < 8; ++r) {
            float v0 = acc0[r] + bias;           // C row = hi*8 + r
            float v1 = acc1[r] + bias;
            sH[hi * 8 + r][ncol]      = v0 > 0.0f ? v0 : 0.0f;   // relu
            sH[16 + hi * 8 + r][ncol] = v1 > 0.0f ? v1 : 0.0f;
        }
    }
    __syncthreads();

    // ---- Layer 2: all 8 waves: (n-tile = wave&3, m-tile = wave>>2) ----
    {
        int ncol  = (wave & 3) * 16 + m;         // 0..63
        int mbase = (wave >> 2) * 16;            // 0 or 16
        v8f acc = {};
        #pragma unroll 4
        for (int kk = 0; kk < HID / 4; ++kk) {
            int k0 = kk * 4 + 2 * hi;
            v2f a, b;
            a.x = sH[mbase + m][k0];
            a.y = sH[mbase + m][k0 + 1];
            b.x = W2[(long long)k0 * OUT_F + ncol];
            b.y = W2[(long long)(k0 + 1) * OUT_F + ncol];
            acc = __builtin_amdgcn_wmma_f32_16x16x4_f32(
                false, a, false, b, (short)0, acc, false, false);
        }
        #pragma unroll
        for (int r = 0; r < 8; ++r) {
            int rr = row0 + mbase + hi * 8 + r;
            if (rr < n_nodes) t[(long long)rr * OUT_F + ncol] = acc[r];
        }
    }
}

// ---------------- launcher ----------------

extern "C" void kernel_launch(void* const* d_in, const int* in_sizes, int n_in,
                              void* d_out, int out_size, void* d_ws, size_t ws_size,
                              hipStream_t stream) {
    (void)n_in; (void)out_size; (void)ws_size;

    const float* x   = (const float*)d_in[0];        // [N,128]
    const int*   ei  = (const int*)d_in[1];          // [2,E] (src row, dst row)
    const float* ew  = (const float*)d_in[2];        // [E]
    const float* W1  = (const float*)d_in[3];        // [128,1024]
    const float* b1  = (const float*)d_in[4];        // [1024]
    const float* W2  = (const float*)d_in[5];        // [1024,64]
    const float* b2  = (const float*)d_in[6];        // [64]
    float* out = (float*)d_out;                      // [N,64]

    const int N = in_sizes[0] / IN_F;                // 100000
    const int E = in_sizes[2];                       // 400000
    const int* src = ei;
    const int* dst = ei + E;

    // workspace layout
    char* ws = (char*)d_ws;
    size_t off = 0;
    float* dinv = (float*)(ws + off); off += ((size_t)N * 4 + 255) & ~(size_t)255;
    float* agg1 = (float*)(ws + off); off += ((size_t)N * IN_F * 4 + 255) & ~(size_t)255;
    float* tbuf = (float*)(ws + off);

    // 1. degrees (self-loop = 1) + edge weights
    k_deg_init<<<(N + 255) / 256, 256, 0, stream>>>(dinv, N);
    k_deg_acc<<<(E + 255) / 256, 256, 0, stream>>>(dst, ew, dinv, E);
    k_dinv<<<(N + 255) / 256, 256, 0, stream>>>(dinv, N);

    // 2. aggregate input features: agg1 = Ā x   (128-wide)
    {
        long long total = (long long)N * IN_F;
        k_agg_init<<<(int)((total + 255) / 256), 256, 0, stream>>>(x, dinv, agg1, N);
        k_edge_agg1<<<(E + 1) / 2, 256, 0, stream>>>(src, dst, ew, dinv, x, agg1, E);
    }

    // 3. fused: t = relu(agg1 @ W1 + b1) @ W2   (WMMA f32, 32-row tiles)
    k_fused_gemm<<<(N + MTILE - 1) / MTILE, 256, 0, stream>>>(agg1, W1, b1, W2, tbuf, N);

    // 4. aggregate outputs: out = Ā t + b2   (64-wide)
    {
        long long total = (long long)N * OUT_F;
        k_out_init<<<(int)((total + 255) / 256), 256, 0, stream>>>(tbuf, dinv, b2, out, N);
        k_edge_agg2<<<(E + 3) / 4, 256, 0, stream>>>(src, dst, ew, dinv, tbuf, out, E);
    }
}